// GaussianBatchLanguageModel_41231686041928
// MI455X (gfx1250) — compile-verified
//
#include <hip/hip_runtime.h>
#include <math.h>

#define DIM   16
#define DIM2  32
#define NTOK  512
#define BB    8
#define LL    64

typedef __attribute__((ext_vector_type(2))) float v2f;
typedef __attribute__((ext_vector_type(8))) float v8f;

// ---------------- Gauss-Jordan helpers (lane-per-row, single wave) ----------------

// 32x65 augmented [A | I | b] -> [I | A^-1 | A^-1 b]
__device__ __forceinline__ void gj32(float aug[DIM2][66], int lane) {
  for (int k = 0; k < DIM2; ++k) {
    float piv = aug[k][k];
    float f = (lane != k) ? aug[lane][k] / piv : 0.0f;
    __syncthreads();
    if (lane != k) {
      for (int j = 0; j < 65; ++j) aug[lane][j] -= f * aug[k][j];
    }
    __syncthreads();
    if (lane == k) {
      float pinv = 1.0f / piv;
      for (int j = 0; j < 65; ++j) aug[k][j] *= pinv;
    }
    __syncthreads();
  }
}

// 16x32 augmented [A | I] -> [I | A^-1]; lanes 16..31 idle but hit barriers
__device__ __forceinline__ void gj16(float aug[DIM][34], int lane) {
  for (int k = 0; k < DIM; ++k) {
    float piv = aug[k][k];
    float f = (lane < DIM && lane != k) ? aug[lane][k] / piv : 0.0f;
    __syncthreads();
    if (lane < DIM && lane != k) {
      for (int j = 0; j < 2 * DIM; ++j) aug[lane][j] -= f * aug[k][j];
    }
    __syncthreads();
    if (lane == k) {
      float pinv = 1.0f / piv;
      for (int j = 0; j < 2 * DIM; ++j) aug[k][j] *= pinv;
    }
    __syncthreads();
  }
}

// ---------------- Kernel 1: lam_t = inv(C^T C), eta_t = lam_t @ mu ----------------
// C^T C is a genuine 32x32x32 f32 GEMM -> 2x2 blocks of 16x16, each as 8 chained
// V_WMMA_F32_16X16X4_F32 (K=32).
__global__ __launch_bounds__(32)
void trans_kernel(const float* __restrict__ tcho, const float* __restrict__ tmu,
                  float* __restrict__ lam_t, float* __restrict__ eta_t) {
  __shared__ float C[DIM2][DIM2 + 1];
  __shared__ float aug[DIM2][66];
  const int r = threadIdx.x;
  const int half = r >> 4;
  const int l16 = r & 15;

  for (int c = 0; c < DIM2; ++c) C[r][c] = tcho[r * DIM2 + c];
  __syncthreads();

  // aug[0:32][0:32] = C^T C
  // A-frag (16x4): lanes 0-15 hold K={kb,kb+1}, lanes 16-31 K={kb+2,kb+3}; M = lane&15.
  // A[m][k] = C[k][bi*16+m]; B[k][n] = C[k][bj*16+n].
#pragma unroll
  for (int bi = 0; bi < 2; ++bi) {
#pragma unroll
    for (int bj = 0; bj < 2; ++bj) {
      v8f acc = {};
#pragma unroll
      for (int kk = 0; kk < 8; ++kk) {
        const int kb = kk * 4 + half * 2;
        v2f a, bfrag;
        a.x = C[kb][bi * DIM + l16];
        a.y = C[kb + 1][bi * DIM + l16];
        bfrag.x = C[kb][bj * DIM + l16];
        bfrag.y = C[kb + 1][bj * DIM + l16];
        acc = __builtin_amdgcn_wmma_f32_16x16x4_f32(
            /*neg_a=*/false, a, /*neg_b=*/false, bfrag,
            /*c_mod=*/(short)0, acc, /*reuse_a=*/false, /*reuse_b=*/false);
      }
#pragma unroll
      for (int q = 0; q < 8; ++q)
        aug[bi * DIM + half * 8 + q][bj * DIM + l16] = acc[q];
    }
  }
  __syncthreads();

  for (int c = 0; c < DIM2; ++c) aug[r][DIM2 + c] = (r == c) ? 1.0f : 0.0f;
  aug[r][64] = tmu[r];
  __syncthreads();
  gj32(aug, r);
  for (int c = 0; c < DIM2; ++c) lam_t[r * DIM2 + c] = aug[r][DIM2 + c];
  eta_t[r] = aug[r][64];
}

// ---------------- Kernel 2: sequential Kalman scan (one wave per batch) ----------------
__global__ __launch_bounds__(32)
void scan_kernel(const int* __restrict__ sentences,
                 const float* __restrict__ emission_mu,
                 const float* __restrict__ emission_cho,
                 const float* __restrict__ lam_t,
                 const float* __restrict__ eta_t,
                 float* __restrict__ pred_mus,
                 float* __restrict__ pred_vars) {
  __shared__ float lamt_s[DIM2][DIM2 + 1];
  __shared__ float etat_s[DIM2];
  __shared__ float aug32[DIM2][66];
  __shared__ float aug16[DIM][34];
  __shared__ float var_s[DIM][DIM + 1];
  __shared__ float pvar_s[DIM][DIM + 1];
  __shared__ float P_s[DIM][DIM + 1];
  __shared__ float mu_s[DIM], wmu_s[DIM], wvar_s[DIM], pmu_s[DIM];

  const int b = blockIdx.x;
  const int lane = threadIdx.x;
  const int half = lane >> 4;
  const int l16 = lane & 15;

  for (int c = 0; c < DIM2; ++c) lamt_s[lane][c] = lam_t[lane * DIM2 + c];
  etat_s[lane] = eta_t[lane];
  if (lane < DIM) {
    mu_s[lane] = 0.0f;
    for (int c = 0; c < DIM; ++c) var_s[lane][c] = (lane == c) ? 1.0f : 0.0f;
  }
  __syncthreads();

  for (int t = 0; t < LL; ++t) {
    const int tok = sentences[b * LL + t];
    if (lane < DIM) {
      wmu_s[lane] = emission_mu[tok * DIM + lane];
      float c = emission_cho[tok * DIM + lane];
      wvar_s[lane] = c * c;
      for (int j = 0; j < DIM; ++j) {
        aug16[lane][j] = var_s[lane][j];
        aug16[lane][DIM + j] = (lane == j) ? 1.0f : 0.0f;
      }
    }
    __syncthreads();
    gj16(aug16, lane);                 // aug16[:,16:32] = lam_in = inv(var)

    float eta_in = 0.0f;
    if (lane < DIM) {
      for (int j = 0; j < DIM; ++j) eta_in += aug16[lane][DIM + j] * mu_s[j];
    }
    // Lam = lam_t + pad(lam_in); augmented with I and eta
    for (int c = 0; c < DIM2; ++c) {
      float v = lamt_s[lane][c];
      if (lane < DIM && c < DIM) v += aug16[lane][DIM + c];
      aug32[lane][c] = v;
      aug32[lane][DIM2 + c] = (lane == c) ? 1.0f : 0.0f;
    }
    aug32[lane][64] = etat_s[lane] + ((lane < DIM) ? eta_in : 0.0f);
    __syncthreads();
    gj32(aug32, lane);                 // cols 32..63 = jv, col 64 = jm

    if (lane < DIM) {
      pmu_s[lane] = aug32[DIM + lane][64];                      // jm[d:]
      for (int j = 0; j < DIM; ++j)
        pvar_s[lane][j] = aug32[DIM + lane][DIM2 + DIM + j];    // jv[d:,d:]
    }
    __syncthreads();

    // ainv = inv(pvar + diag(wvar))
    if (lane < DIM) {
      for (int j = 0; j < DIM; ++j) {
        aug16[lane][j] = pvar_s[lane][j] + ((lane == j) ? wvar_s[lane] : 0.0f);
        aug16[lane][DIM + j] = (lane == j) ? 1.0f : 0.0f;
      }
    }
    __syncthreads();
    gj16(aug16, lane);                 // aug16[:,16:32] = ainv

    // P = pvar @ ainv : 16x16x16 f32 GEMM as 4 chained V_WMMA_F32_16X16X4_F32.
    v8f acc = {};
#pragma unroll
    for (int kk = 0; kk < 4; ++kk) {
      const int kb = kk * 4 + half * 2;
      v2f a, bfrag;
      a.x = pvar_s[l16][kb];
      a.y = pvar_s[l16][kb + 1];
      bfrag.x = aug16[kb][DIM + l16];
      bfrag.y = aug16[kb + 1][DIM + l16];
      acc = __builtin_amdgcn_wmma_f32_16x16x4_f32(
          /*neg_a=*/false, a, /*neg_b=*/false, bfrag,
          /*c_mod=*/(short)0, acc, /*reuse_a=*/false, /*reuse_b=*/false);
    }
#pragma unroll
    for (int r = 0; r < 8; ++r) P_s[half * 8 + r][l16] = acc[r];
    __syncthreads();

    if (lane < DIM) {
      float tv = 0.0f;
      for (int j = 0; j < DIM; ++j) tv += aug16[lane][DIM + j] * pmu_s[j];   // ainv@pmu
      float pmv = 0.0f;
      for (int j = 0; j < DIM; ++j) pmv += P_s[lane][j] * wmu_s[j];          // P@wmu
      const float nmu = wvar_s[lane] * tv + pmv;
      pred_mus[(b * LL + t) * DIM + lane] = nmu;
      for (int j = 0; j < DIM; ++j) {
        const float nv = P_s[lane][j] * wvar_s[j];
        pred_vars[((b * LL + t) * DIM + lane) * DIM + j] = nv;
        var_s[lane][j] = nv;
      }
      mu_s[lane] = nmu;
    }
    __syncthreads();
  }
}

// ---------------- Kernel 3: per-token 16x16 Cholesky scoring ----------------
#define TI(i, j) ((i) * ((i) + 1) / 2 + (j))

__global__ __launch_bounds__(256)
void score_kernel(const float* __restrict__ pred_mus,
                  const float* __restrict__ pred_vars,
                  const float* __restrict__ decoder_mu,
                  const float* __restrict__ decoder_cho,
                  float* __restrict__ out) {
  __shared__ float pv[DIM][DIM];
  __shared__ float pm[DIM];
  const int bl = blockIdx.y;                       // b*L + l
  const int n = blockIdx.x * blockDim.x + threadIdx.x;

  if (threadIdx.x < DIM * DIM)
    ((float*)pv)[threadIdx.x] = pred_vars[bl * DIM * DIM + threadIdx.x];
  if (threadIdx.x < DIM)
    pm[threadIdx.x] = pred_mus[bl * DIM + threadIdx.x];
  __syncthreads();

  float dv[DIM];
#pragma unroll
  for (int i = 0; i < DIM; ++i) {
    float c = decoder_cho[n * DIM + i];
    dv[i] = c * c;
  }

  float Lm[TI(15, 15) + 1];   // 136-entry lower triangle; constant-indexed -> registers
#pragma unroll
  for (int j = 0; j < DIM; ++j) {
    float s = pv[j][j] + dv[j];
#pragma unroll
    for (int k = 0; k < j; ++k) s -= Lm[TI(j, k)] * Lm[TI(j, k)];
    const float d = sqrtf(s);
    Lm[TI(j, j)] = d;
    const float rd = 1.0f / d;
#pragma unroll
    for (int i = j + 1; i < DIM; ++i) {
      float s2 = pv[i][j];
#pragma unroll
      for (int k = 0; k < j; ++k) s2 -= Lm[TI(i, k)] * Lm[TI(j, k)];
      Lm[TI(i, j)] = s2 * rd;
    }
  }

  float sol[DIM];
  float quad = 0.0f, logdet = 0.0f;
#pragma unroll
  for (int i = 0; i < DIM; ++i) {
    float s = pm[i] - decoder_mu[n * DIM + i];
#pragma unroll
    for (int k = 0; k < i; ++k) s -= Lm[TI(i, k)] * sol[k];
    const float di = Lm[TI(i, i)];
    s /= di;
    sol[i] = s;
    quad += s * s;
    logdet += __logf(di);
  }
  logdet *= 2.0f;
  const float LOG2PI = 1.8378770664093453f;
  out[bl * NTOK + n] = -0.5f * ((float)DIM * LOG2PI + logdet + quad);
}

// ---------------- Launch ----------------
extern "C" void kernel_launch(void* const* d_in, const int* in_sizes, int n_in,
                              void* d_out, int out_size, void* d_ws, size_t ws_size,
                              hipStream_t stream) {
  (void)in_sizes; (void)n_in; (void)out_size; (void)ws_size;
  const int*   sentences      = (const int*)d_in[0];
  // d_in[1] = masks (all ones; unused by the reference math)
  const float* emission_mu    = (const float*)d_in[2];
  const float* emission_cho   = (const float*)d_in[3];
  const float* transition_mu  = (const float*)d_in[4];
  const float* transition_cho = (const float*)d_in[5];
  const float* decoder_mu     = (const float*)d_in[6];
  const float* decoder_cho    = (const float*)d_in[7];
  float* out = (float*)d_out;

  float* ws        = (float*)d_ws;
  float* lam_t     = ws;                                   // 32*32
  float* eta_t     = lam_t + DIM2 * DIM2;                  // 32
  float* pred_mus  = eta_t + DIM2;                         // B*L*16
  float* pred_vars = pred_mus + BB * LL * DIM;             // B*L*16*16  (~550 KB total)

  trans_kernel<<<1, 32, 0, stream>>>(transition_cho, transition_mu, lam_t, eta_t);
  scan_kernel<<<BB, 32, 0, stream>>>(sentences, emission_mu, emission_cho,
                                     lam_t, eta_t, pred_mus, pred_vars);
  dim3 grid(NTOK / 256, BB * LL);
  score_kernel<<<grid, 256, 0, stream>>>(pred_mus, pred_vars, decoder_mu, decoder_cho, out);
}